// SAGEEncoder_48533130445494
// MI455X (gfx1250) — compile-verified
//
#include <hip/hip_runtime.h>

typedef __attribute__((ext_vector_type(2))) float v2f;
typedef __attribute__((ext_vector_type(8))) float v8f;

#define NNODES 50000
#define NEDGES 600000
#define FEAT   128
#define LDS_STRIDE 132  // dword row stride: (132*i + k) % 64 == (4*i + k) % 64
                        // -> 16 rows x 4 consecutive K dwords hit 64 distinct banks

// ---------------------------------------------------------------- zero fill
__global__ void zero_f32(float* __restrict__ p, long n) {
  long i = (long)blockIdx.x * blockDim.x + threadIdx.x;
  long stride = (long)gridDim.x * blockDim.x;
  for (; i < n; i += stride) p[i] = 0.0f;
}

// ------------------------------------------------- edge scatter (mean accum)
// One wave per edge: 32 lanes x float4 = 128 floats. Atomic-add into agg[dst].
__global__ void scatter_mean_accum(const float* __restrict__ h,
                                   const long long* __restrict__ edge,
                                   float* __restrict__ agg,
                                   float* __restrict__ cnt,
                                   int nedges) {
  int wavesPerBlock = blockDim.x >> 5;
  int e = blockIdx.x * wavesPerBlock + ((int)threadIdx.x >> 5);
  int lane = threadIdx.x & 31;
  if (e >= nedges) return;
  int src = (int)edge[e];
  int dst = (int)edge[nedges + e];
  const float4* hs = (const float4*)(h + (long)src * FEAT);
  float4 v = hs[lane];
  float* a = agg + (long)dst * FEAT + lane * 4;
  atomicAdd(a + 0, v.x);
  atomicAdd(a + 1, v.y);
  atomicAdd(a + 2, v.z);
  atomicAdd(a + 3, v.w);
  if (cnt != nullptr && lane == 0) atomicAdd(cnt + dst, 1.0f);
}

// ------------------------------------------------------------- mean divide
__global__ void finalize_mean(float* __restrict__ agg,
                              const float* __restrict__ cnt, int nrows) {
  long i = (long)blockIdx.x * blockDim.x + threadIdx.x;
  long n = (long)nrows * FEAT;
  if (i >= n) return;
  int row = (int)(i >> 7);  // FEAT == 128
  agg[i] *= 1.0f / fmaxf(cnt[row], 1.0f);
}

// --------------------------------------------------- fused SAGE layer GEMM
// out[m,:] = relu( Agg[m,:] @ Wl + Hs[m,:] @ Wr + bias )   (full f32 WMMA)
// Block = 256 threads (8 waves). Block owns a 16-row M tile; wave w owns
// output columns [16w, 16w+16). Weights staged TRANSPOSED in LDS so each
// lane's (k, k+1) B-pair is one ds_load_b64; A tiles likewise. All hot-loop
// DS addresses are base-pointer + immediate-k offsets (no per-step VALU).
__global__ __launch_bounds__(256) void sage_gemm_wmma(
    const float* __restrict__ Agg,   // [N,128] mean-aggregated neighbors
    const float* __restrict__ Hs,    // [N,128] self/root features
    const float* __restrict__ Wl,
    const float* __restrict__ Wr,
    const float* __restrict__ bias,
    float* __restrict__ out,
    const long long* __restrict__ out_idx)  // optional row remap (pos_idx)
{
  __shared__ float sA[16 * LDS_STRIDE];          //  8448 B
  __shared__ float sH[16 * LDS_STRIDE];          //  8448 B
  __shared__ float sWlT[FEAT * LDS_STRIDE];      // 67584 B (transposed: [col][k])
  __shared__ float sWrT[FEAT * LDS_STRIDE];      // 67584 B -> total ~148.5 KB

  int t = threadIdx.x;
  // Stage weights transposed: sWT[c][r] = W[r][c]. 16 float4 rows per thread.
  for (int i = t; i < FEAT * (FEAT / 4); i += 256) {
    int r = i >> 5;
    int c = (i & 31) * 4;
    float4 v = *(const float4*)(Wl + r * FEAT + c);
    sWlT[(c + 0) * LDS_STRIDE + r] = v.x;
    sWlT[(c + 1) * LDS_STRIDE + r] = v.y;
    sWlT[(c + 2) * LDS_STRIDE + r] = v.z;
    sWlT[(c + 3) * LDS_STRIDE + r] = v.w;
    v = *(const float4*)(Wr + r * FEAT + c);
    sWrT[(c + 0) * LDS_STRIDE + r] = v.x;
    sWrT[(c + 1) * LDS_STRIDE + r] = v.y;
    sWrT[(c + 2) * LDS_STRIDE + r] = v.z;
    sWrT[(c + 3) * LDS_STRIDE + r] = v.w;
  }
  // Stage the 16 A-rows of both input matrices (row-major, padded stride).
  long m0 = (long)blockIdx.x * 16;
  for (int i = t; i < 16 * (FEAT / 4); i += 256) {
    int r = i >> 5;
    int c = (i & 31) * 4;
    float4 v = *(const float4*)(Agg + (m0 + r) * FEAT + c);
    float* d = sA + r * LDS_STRIDE + c;
    d[0] = v.x; d[1] = v.y; d[2] = v.z; d[3] = v.w;
    v = *(const float4*)(Hs + (m0 + r) * FEAT + c);
    d = sH + r * LDS_STRIDE + c;
    d[0] = v.x; d[1] = v.y; d[2] = v.z; d[3] = v.w;
  }
  __syncthreads();

  int wave = t >> 5;
  int lane = t & 31;
  int row  = lane & 15;          // A-matrix row (M) held by this lane
  int kh   = (lane >> 4) << 1;   // K sub-offset: lanes 16-31 hold K+2,K+3
  int col  = wave * 16 + (lane & 15);  // output column (N)

  // Hot-loop base pointers (computed once; loop adds immediate k*4 only).
  const float* aP = sA   + row * LDS_STRIDE + kh;
  const float* hP = sH   + row * LDS_STRIDE + kh;
  const float* lP = sWlT + col * LDS_STRIDE + kh;
  const float* rP = sWrT + col * LDS_STRIDE + kh;

  // Two independent accumulator chains -> 2x WMMA ILP.
  v8f accL = {};
  v8f accR = {};
#pragma unroll
  for (int k = 0; k < FEAT; k += 4) {
    v2f a  = *(const v2f*)(aP + k);
    v2f bl = *(const v2f*)(lP + k);
    accL = __builtin_amdgcn_wmma_f32_16x16x4_f32(false, a, false, bl,
                                                 (short)0, accL, false, false);
    v2f h  = *(const v2f*)(hP + k);
    v2f br = *(const v2f*)(rP + k);
    accR = __builtin_amdgcn_wmma_f32_16x16x4_f32(false, h, false, br,
                                                 (short)0, accR, false, false);
  }

  float bv = bias[col];
  long r0 = m0 + ((lane >> 4) << 3);  // D rows: lanes 16-31 hold M=8..15
#pragma unroll
  for (int r = 0; r < 8; ++r) {
    float v = fmaxf(accL[r] + accR[r] + bv, 0.0f);  // bias + relu
    long grow = r0 + r;
    long orow = out_idx ? out_idx[grow] : grow;
    out[orow * FEAT + col] = v;
  }
}

// ---------------------------------------------------------------- launcher
extern "C" void kernel_launch(void* const* d_in, const int* in_sizes, int n_in,
                              void* d_out, int out_size, void* d_ws, size_t ws_size,
                              hipStream_t stream) {
  const float*     x       = (const float*)d_in[0];
  const long long* edge    = (const long long*)d_in[1];
  const long long* pos_idx = (const long long*)d_in[2];
  // d_in[3] = pad_n scalar (unused; out_size covers it)
  const float* Wl1 = (const float*)d_in[4];
  const float* Wr1 = (const float*)d_in[5];
  const float* b1  = (const float*)d_in[6];
  const float* Wl2 = (const float*)d_in[7];
  const float* Wr2 = (const float*)d_in[8];
  const float* b2  = (const float*)d_in[9];
  float* out = (float*)d_out;

  // Workspace layout: agg[N*128] | h1[N*128] | cnt[N]   (~51.4 MB)
  float* agg = (float*)d_ws;
  float* h1  = agg + (long)NNODES * FEAT;
  float* cnt = h1 + (long)NNODES * FEAT;

  const long naggr = (long)NNODES * FEAT;
  const int scatterBlocks = (NEDGES + 7) / 8;     // 8 waves (edges) / block
  const int finBlocks = (int)((naggr + 255) / 256);
  const int gemmBlocks = NNODES / 16;             // 3125, exact

  // ---- layer 1 ----
  zero_f32<<<2048, 256, 0, stream>>>(agg, naggr);
  zero_f32<<<64,   256, 0, stream>>>(cnt, NNODES);
  zero_f32<<<2048, 256, 0, stream>>>(out, (long)out_size);

  scatter_mean_accum<<<scatterBlocks, 256, 0, stream>>>(x, edge, agg, cnt, NEDGES);
  finalize_mean<<<finBlocks, 256, 0, stream>>>(agg, cnt, NNODES);
  sage_gemm_wmma<<<gemmBlocks, 256, 0, stream>>>(agg, x, Wl1, Wr1, b1, h1, nullptr);

  // ---- layer 2 (counts identical: same edge list) ----
  zero_f32<<<2048, 256, 0, stream>>>(agg, naggr);
  scatter_mean_accum<<<scatterBlocks, 256, 0, stream>>>(h1, edge, agg, nullptr, NEDGES);
  finalize_mean<<<finBlocks, 256, 0, stream>>>(agg, cnt, NNODES);
  sage_gemm_wmma<<<gemmBlocks, 256, 0, stream>>>(agg, h1, Wl2, Wr2, b2, out, pos_idx);
}